// HardBootstrappingLoss_59992103190991
// MI455X (gfx1250) — compile-verified
//
#include <hip/hip_runtime.h>
#include <hip/hip_bf16.h>
#include <stdint.h>

#define B_ROWS 4096
#define C_COLS 50257
#define BDIM   512
#define FULL_TILES 24                       // floor(((C_COLS - head)/4)/BDIM) for head in 0..3
#define VEC_ELEMS  (FULL_TILES * BDIM * 4)  // 49152 elements covered by the vector tiles

// gfx1250 async global->LDS path (guarded: host pass / older toolchains fall back)
#if defined(__gfx1250__) && __has_builtin(__builtin_amdgcn_global_load_async_to_lds_b128) && __has_builtin(__builtin_amdgcn_s_wait_asynccnt)
#define HAVE_ASYNC 1
#else
#define HAVE_ASYNC 0
#endif

typedef int v4i __attribute__((ext_vector_type(4)));

__device__ __forceinline__ void async_cp16(const float* g, float4* l) {
#if HAVE_ASYNC
  __builtin_amdgcn_global_load_async_to_lds_b128(
      (__attribute__((address_space(1))) v4i*)(g),
      (__attribute__((address_space(3))) v4i*)(l),
      0, 0);
#else
  *l = *(const float4*)g;
#endif
}

__device__ __forceinline__ void async_wait1() {
#if HAVE_ASYNC
  __builtin_amdgcn_s_wait_asynccnt(1);   // oldest copy done (in-order completion)
#endif
  __asm__ volatile("" ::: "memory");
}

__device__ __forceinline__ void async_wait0() {
#if HAVE_ASYNC
  __builtin_amdgcn_s_wait_asynccnt(0);
#endif
  __asm__ volatile("" ::: "memory");
}

__global__ __launch_bounds__(BDIM) void
hb_rowstat_kernel(const float* __restrict__ y_pred, const int* __restrict__ y,
                  float* __restrict__ out, float* __restrict__ wsc,
                  int* __restrict__ wss)
{
  __shared__ float4 stage[2][BDIM];   // 16 KB async staging (double buffer)
  __shared__ float  rm[BDIM];
  __shared__ float  rs[BDIM];
  __shared__ float  tv[BDIM][3];
  __shared__ int    tix[BDIM][3];

  const int tid = threadIdx.x;
  const int row = blockIdx.x;
  const float* rp = y_pred + (size_t)row * C_COLS;

  float xy = 0.0f;
  if (tid == 0) xy = rp[y[row]];

  // online softmax state + running top-3 (value, index) per thread
  float m = -3.0e38f, zsum = 0.0f;
  float v1 = -3.0e38f, v2 = -3.0e38f, v3 = -3.0e38f;
  int   i1 = -1, i2 = -1, i3 = -1;

  auto ins = [&](float v, int i) {
    if ((v > v3) || (v == v3 && i < i3)) {
      if ((v > v1) || (v == v1 && i < i1))      { v3=v2; i3=i2; v2=v1; i2=i1; v1=v; i1=i; }
      else if ((v > v2) || (v == v2 && i < i2)) { v3=v2; i3=i2; v2=v;  i2=i; }
      else                                      { v3=v;  i3=i; }
    }
  };
  auto acc1 = [&](float x, int idx) {
    if (x > m) { zsum *= __expf(m - x); m = x; }
    zsum += __expf(x - m);
    ins(x, idx);
  };
  auto acc4 = [&](float4 q, int base) {
    float m4 = fmaxf(fmaxf(q.x, q.y), fmaxf(q.z, q.w));
    if (m4 > m) { zsum *= __expf(m - m4); m = m4; }   // rare rescale
    zsum += __expf(q.x - m) + __expf(q.y - m) + __expf(q.z - m) + __expf(q.w - m);
    ins(q.x, base); ins(q.y, base + 1); ins(q.z, base + 2); ins(q.w, base + 3);
  };

  // Rows are only 4B aligned (C_COLS % 4 == 1): peel 0..3 head elements so all
  // b128 global/async accesses are 16B aligned.
  const int head = (int)((4u - ((((uintptr_t)rp) >> 2) & 3u)) & 3u);
  for (int idx = tid; idx < head; idx += BDIM) acc1(rp[idx], idx);

  const float* vbase = rp + head;

  // Double-buffered async global->LDS pipeline over 24 full tiles of 512 float4.
  async_cp16(vbase + (size_t)tid * 4, &stage[0][tid]);
  #pragma unroll 1
  for (int t = 0; t < FULL_TILES - 1; ++t) {
    async_cp16(vbase + ((size_t)(t + 1) * BDIM + tid) * 4, &stage[(t + 1) & 1][tid]);
    async_wait1();                       // tile t resident in LDS
    float4 q = stage[t & 1][tid];        // ds_load_b128 (own lane's data; per-wave sync only)
    acc4(q, head + (t * BDIM + tid) * 4);
  }
  async_wait0();
  {
    float4 q = stage[(FULL_TILES - 1) & 1][tid];
    acc4(q, head + ((FULL_TILES - 1) * BDIM + tid) * 4);
  }

  // Scalar tail (~1105 elements).
  for (int idx = head + VEC_ELEMS + tid; idx < C_COLS; idx += BDIM) acc1(rp[idx], idx);

  // Block reduction: merge (m, Z) with exp-rescale; merge top-3 triples.
  rm[tid] = m; rs[tid] = zsum;
  tv[tid][0] = v1; tv[tid][1] = v2; tv[tid][2] = v3;
  tix[tid][0] = i1; tix[tid][1] = i2; tix[tid][2] = i3;
  __syncthreads();
  for (int off = BDIM >> 1; off > 0; off >>= 1) {
    if (tid < off) {
      float mo = rm[tid + off], so = rs[tid + off];
      float mm = fmaxf(m, mo);
      zsum = zsum * __expf(m - mm) + so * __expf(mo - mm);
      m = mm;
      rm[tid] = m; rs[tid] = zsum;
      ins(tv[tid + off][0], tix[tid + off][0]);
      ins(tv[tid + off][1], tix[tid + off][1]);
      ins(tv[tid + off][2], tix[tid + off][2]);
      tv[tid][0] = v1; tv[tid][1] = v2; tv[tid][2] = v3;
      tix[tid][0] = i1; tix[tid][1] = i2; tix[tid][2] = i3;
    }
    __syncthreads();
  }

  if (tid == 0) {
    float logZ = __logf(zsum);
    float mz = m + logZ;                       // -logp[j] = mz - x[j]
    float p1 = __expf(v1 - m) / zsum;
    float p2 = __expf(v2 - m) / zsum;          // max_9 = second-largest prob
    float p3 = __expf(v3 - m) / zsum;
    float fy = __expf(xy - m) / zsum;
    int   s  = (fy < 0.02f) ? 1 : 0;
    float contrib;
    if (s) {
      float ws = p1 + p2 + p3;
      contrib = (p1 * (mz - v1) + p2 * (mz - v2) + p3 * (mz - v3)) / ws;
    } else {
      contrib = mz - xy;
    }
    // tuple flattened as float: [boot][s×B][z×B][max9_d×B][max8_d×B][count][s*max_9×B]
    out[1 + row]                = (float)s;
    out[1 + B_ROWS + row]       = s ? (float)i1 : -1.0f;   // s*argmax + (1-s)*-1
    out[1 + 2 * B_ROWS + row]   = s ? (float)i2 : -1.0f;
    out[1 + 3 * B_ROWS + row]   = s ? (float)i3 : -1.0f;
    out[2 + 4 * B_ROWS + row]   = s ? p2 : 0.0f;
    wsc[row] = contrib;
    wss[row] = s;
  }
}

// Deterministic final reduction (fixed-order tree; no float atomics).
__global__ __launch_bounds__(BDIM) void
hb_finalize_kernel(const float* __restrict__ wsc, const int* __restrict__ wss,
                   float* __restrict__ out)
{
  __shared__ float sb[BDIM];
  __shared__ int   sc[BDIM];
  const int tid = threadIdx.x;
  float acc = 0.0f; int cnt = 0;
  for (int i = tid; i < B_ROWS; i += BDIM) { acc += wsc[i]; cnt += 1 - wss[i]; }
  sb[tid] = acc; sc[tid] = cnt;
  __syncthreads();
  for (int off = BDIM >> 1; off > 0; off >>= 1) {
    if (tid < off) { sb[tid] += sb[tid + off]; sc[tid] += sc[tid + off]; }
    __syncthreads();
  }
  if (tid == 0) {
    out[0]              = sb[0] / (float)B_ROWS;   // bootstrap
    out[1 + 4 * B_ROWS] = (float)sc[0];            // sum(1 - s)
  }
}

extern "C" void kernel_launch(void* const* d_in, const int* in_sizes, int n_in,
                              void* d_out, int out_size, void* d_ws, size_t ws_size,
                              hipStream_t stream) {
  (void)in_sizes; (void)n_in; (void)out_size; (void)ws_size;
  const float* y_pred = (const float*)d_in[0];
  const int*   y      = (const int*)d_in[1];
  float* out = (float*)d_out;
  float* wsc = (float*)d_ws;                                   // 4096 floats
  int*   wss = (int*)((char*)d_ws + B_ROWS * sizeof(float));   // 4096 ints

  hb_rowstat_kernel<<<B_ROWS, BDIM, 0, stream>>>(y_pred, y, out, wsc, wss);
  hb_finalize_kernel<<<1, BDIM, 0, stream>>>(wsc, wss, out);
}